// GridSamplePScan_45913200394381
// MI455X (gfx1250) — compile-verified
//
#include <hip/hip_runtime.h>
#include <hip/hip_bf16.h>
#include <stdint.h>

typedef __attribute__((ext_vector_type(2))) float v2f;
typedef __attribute__((ext_vector_type(8))) float v8f;
typedef __attribute__((ext_vector_type(4))) unsigned int v4u;
typedef __attribute__((ext_vector_type(4))) int v4i;
typedef __attribute__((ext_vector_type(8))) int v8i;

#define LL 64
#define CH 8
#define HW 4096          // 64*64 pixels
#define PLANE 8192       // 2*HW floats of flow per (b,l)
#define IMG_ELEMS 32768  // CH*HW floats per (b,l) image
#define IMG_BYTES 131072 // IMG_ELEMS*4

__device__ __forceinline__ int clampi(int v, int lo, int hi) {
  return v < lo ? lo : (v > hi ? hi : v);
}

// ---------------------------------------------------------------------------
// Kernel 1: cum[b,:,c,h,w] = cumsum_L(flows) == tril(ones(64,64)) @ X(64x8192)
// per batch, done with V_WMMA_F32_16X16X4_F32. One wave per 16x16 output tile.
// A (16x4, triangular constant) is synthesized in registers; B (4x16) is
// loaded per the 32-bit operand layout (VGPR0 = K0 | K2 halves, VGPR1 = K1|K3).
// ---------------------------------------------------------------------------
__global__ __launch_bounds__(32) void cumsum_wmma(const float* __restrict__ flows,
                                                  float* __restrict__ cum) {
  const int j0 = blockIdx.x * 16;  // column tile within 8192
  const int m0 = blockIdx.y * 16;  // L-row tile
  const int b  = blockIdx.z;
  const int lane = threadIdx.x & 31;
  const int hf = lane >> 4;   // which half of the wave
  const int ln = lane & 15;
  const int m  = m0 + ln;     // output row this lane produces (A-matrix M)

  const float* X = flows + (size_t)b * (LL * PLANE);
  v8f acc = {};
  // Triangular: only k <= m0+15 contributes.
  for (int k0 = 0; k0 <= m0 + 15; k0 += 4) {
    const int ka = k0 + hf * 2;     // K index for .x component
    const int kb = ka + 1;          // K index for .y component
    v2f av, bv;
    av.x = (ka <= m) ? 1.0f : 0.0f; // tril(ones)
    av.y = (kb <= m) ? 1.0f : 0.0f;
    bv.x = X[(size_t)ka * PLANE + j0 + ln];
    bv.y = X[(size_t)kb * PLANE + j0 + ln];
    acc = __builtin_amdgcn_wmma_f32_16x16x4_f32(false, av, false, bv,
                                                (short)0, acc, false, false);
  }
  float* Y = cum + (size_t)b * (LL * PLANE);
  const int rbase = m0 + hf * 8;    // C/D layout: VGPR r -> M=r (lanes 0-15), M=r+8 (16-31)
#pragma unroll
  for (int r = 0; r < 8; ++r)
    Y[(size_t)(rbase + r) * PLANE + j0 + ln] = acc[r];
}

// ---------------------------------------------------------------------------
// TDM: DMA one (b,k) image slice (128KB, 32768 f32, 1-D tile) into LDS.
// D# built per CDNA5 ISA 8.3/8.4 bit layout. Issued by wave 0 only; tracked
// with TENSORcnt. 6-arg builtin form (clang-23 toolchain).
// ---------------------------------------------------------------------------
__device__ __forceinline__ void tdm_load_image(const float* gsrc, unsigned lds_byte_addr) {
  const uint64_t ga = (uint64_t)(uintptr_t)gsrc;
  v4u g0;
  g0.x = 1u;                                   // count=1, is_restore=0, gather off
  g0.y = lds_byte_addr;                        // lds_addr [63:32]
  g0.z = (unsigned)(ga & 0xFFFFFFFFu);         // global_addr [95:64]
  g0.w = (unsigned)((ga >> 32) & 0x01FFFFFFu)  // global_addr [120:96]
         | 0x80000000u;                        // type=2 ("image") in [127:126]
  v8i g1;
  g1[0] = (int)(2u << 16);                                 // data_size=4B; wg_mask=0
  g1[1] = (int)(((unsigned)IMG_ELEMS & 0xFFFFu) << 16);    // tensor_dim0 lo16 @ [63:48]
  g1[2] = (int)((((unsigned)IMG_ELEMS >> 16) & 0xFFFFu)    // tensor_dim0 hi16 @ [79:64]
         | (1u << 16));                                    // tensor_dim1=1 @ [111:80]
  g1[3] = (int)(((unsigned)IMG_ELEMS & 0xFFFFu) << 16);    // tile_dim0 @ [127:112]
  g1[4] = 1;                                               // tile_dim1=1, tile_dim2=0
  g1[5] = (int)IMG_ELEMS;                                  // tensor_dim0_stride lo32
  g1[6] = (int)(((unsigned)IMG_ELEMS & 0xFFFFu) << 16);    // stride0 hi16=0 | stride1 lo16
  g1[7] = (int)(((unsigned)IMG_ELEMS >> 16) & 0xFFFFu);    // tensor_dim1_stride hi
  v4i g2 = {0, 0, 0, 0};
  v4i g3 = {0, 0, 0, 0};
  v8i g4 = {0, 0, 0, 0, 0, 0, 0, 0};
  __builtin_amdgcn_tensor_load_to_lds(g0, g1, g2, g3, g4, 0);
}

// ---------------------------------------------------------------------------
// Kernel 2: one workgroup per (b,t). Loop k=0..t, double-buffering images[b,k]
// in LDS via the Tensor Data Mover, bilinear-gather 4 taps x 8 channels per
// pixel from LDS, accumulate in registers. 1024 threads, 4 pixels/thread,
// 32 f32 accumulators/thread.
// ---------------------------------------------------------------------------
__global__ __launch_bounds__(1024) void warp_accum(const float* __restrict__ images,
                                                   const float* __restrict__ cum,
                                                   float* __restrict__ out) {
  extern __shared__ float smem[];  // 2 * IMG_ELEMS floats (256KB dynamic LDS)
  const int t = blockIdx.x;
  const int b = blockIdx.y;
  const int tid = threadIdx.x;

  // Dynamic LDS starts right after static LDS; no pointer-cast gymnastics.
  const unsigned lds_base = (unsigned)__builtin_amdgcn_groupstaticsize();

  const float* cumB = cum + (size_t)b * (LL * PLANE);
  const float* ft = cumB + (size_t)t * PLANE;

  int   px[4];
  float bx[4], by[4], ftx[4], fty[4];
#pragma unroll
  for (int i = 0; i < 4; ++i) {
    const int p = tid + i * 1024;
    px[i] = p;
    const int h = p >> 6, w = p & 63;
    bx[i] = (w + 0.5f) * (1.0f / 32.0f) - 1.0f;   // base grid x
    by[i] = (h + 0.5f) * (1.0f / 32.0f) - 1.0f;   // base grid y
    ftx[i] = ft[p];
    fty[i] = ft[HW + p];
  }

  float acc[4][CH];
#pragma unroll
  for (int i = 0; i < 4; ++i)
#pragma unroll
    for (int c = 0; c < CH; ++c) acc[i][c] = 0.0f;

  const float* imgB = images + (size_t)b * (size_t)(LL * IMG_ELEMS);

  if (tid < 32) tdm_load_image(imgB, lds_base);  // prefetch k=0 -> buffer 0

  for (int k = 0; k <= t; ++k) {
    if (tid < 32) __builtin_amdgcn_s_wait_tensorcnt(0);  // buffer (k&1) landed
    __syncthreads();
    if (k < t && tid < 32)                               // prefetch k+1
      tdm_load_image(imgB + (size_t)(k + 1) * IMG_ELEMS,
                     lds_base + ((unsigned)(k + 1) & 1u) * IMG_BYTES);

    const float* img = smem + (size_t)(k & 1) * IMG_ELEMS;
    const float* fk = cumB + (size_t)k * PLANE;
#pragma unroll
    for (int i = 0; i < 4; ++i) {
      const int p = px[i];
      float gx = bx[i] + ftx[i] - fk[p];
      const float gy = by[i] + fty[i] - fk[HW + p];
      // x wrap: remainder(gx+1, 2) - 1  (positive remainder)
      const float gx1 = gx + 1.0f;
      gx = gx1 - 2.0f * floorf(gx1 * 0.5f) - 1.0f;
      const float ix = (gx + 1.0f) * 32.0f - 0.5f;  // ((gx+1)*64 - 1)*0.5
      const float iy = (gy + 1.0f) * 32.0f - 0.5f;
      const float x0f = floorf(ix), y0f = floorf(iy);
      const float wx1 = ix - x0f, wx0 = 1.0f - wx1;
      const float wy1 = iy - y0f, wy0 = 1.0f - wy1;
      const int x0 = (int)x0f, y0 = (int)y0f;
      const int x1 = x0 + 1, y1 = y0 + 1;
      const float vx0 = (x0 >= 0 && x0 < 64) ? 1.0f : 0.0f;
      const float vx1 = (x1 >= 0 && x1 < 64) ? 1.0f : 0.0f;
      const float vy0 = (y0 >= 0 && y0 < 64) ? 1.0f : 0.0f;
      const float vy1 = (y1 >= 0 && y1 < 64) ? 1.0f : 0.0f;
      const float w00 = wy0 * wx0 * vy0 * vx0;
      const float w01 = wy0 * wx1 * vy0 * vx1;
      const float w10 = wy1 * wx0 * vy1 * vx0;
      const float w11 = wy1 * wx1 * vy1 * vx1;
      const int x0c = clampi(x0, 0, 63), x1c = clampi(x1, 0, 63);
      const int y0c = clampi(y0, 0, 63), y1c = clampi(y1, 0, 63);
      const int o00 = y0c * 64 + x0c, o01 = y0c * 64 + x1c;
      const int o10 = y1c * 64 + x0c, o11 = y1c * 64 + x1c;
#pragma unroll
      for (int c = 0; c < CH; ++c) {
        const float* chp = img + c * HW;
        acc[i][c] = fmaf(w00, chp[o00],
                    fmaf(w01, chp[o01],
                    fmaf(w10, chp[o10],
                    fmaf(w11, chp[o11], acc[i][c]))));
      }
    }
    __syncthreads();  // everyone done reading buffer (k&1) before it is refilled
  }

  float* outp = out + (size_t)(b * LL + t) * (size_t)(CH * HW);
#pragma unroll
  for (int c = 0; c < CH; ++c)
#pragma unroll
    for (int i = 0; i < 4; ++i)
      outp[c * HW + px[i]] = acc[i][c];
}

extern "C" void kernel_launch(void* const* d_in, const int* in_sizes, int n_in,
                              void* d_out, int out_size, void* d_ws, size_t ws_size,
                              hipStream_t stream) {
  const float* flows  = (const float*)d_in[0];   // (4,64,2,64,64) f32
  const float* images = (const float*)d_in[1];   // (4,64,8,64,64) f32
  float* out = (float*)d_out;                    // (4,64,8,64,64) f32
  float* cum = (float*)d_ws;                     // 8.4 MB scratch for cumsum

  // Allow 256KB dynamic LDS (double-buffered 128KB image slices).
  (void)hipFuncSetAttribute((const void*)warp_accum,
                            hipFuncAttributeMaxDynamicSharedMemorySize,
                            2 * IMG_BYTES);

  dim3 g1(512, 4, 4);                    // 8192/16 col tiles, 4 row tiles, 4 batches
  cumsum_wmma<<<g1, 32, 0, stream>>>(flows, cum);

  dim3 g2(64, 4, 1);                     // (t, b)
  warp_accum<<<g2, 1024, 2 * IMG_BYTES, stream>>>(images, cum, out);
}